// NTXentLoss_29661044146918
// MI455X (gfx1250) — compile-verified
//
#include <hip/hip_runtime.h>
#include <hip/hip_bf16.h>
#include <math.h>

typedef float v2f __attribute__((ext_vector_type(2)));
typedef float v8f __attribute__((ext_vector_type(8)));

#define BATCH 4096
#define DIM 256
#define ROWS_PER_BLOCK 16
#define A_STRIDE 260   // 256 + 4 pad: ds_load_b64 of A operand hits all 64 banks
#define INV_T 2.0f     // 1 / temperature (0.5)

// d_ws accumulators: [0]=loss sum (f32), [1]=valid anchor count (u32)
__global__ void ntxent_init(float* acc) {
    acc[0] = 0.0f;
    ((unsigned int*)acc)[1] = 0u;
}

__global__ __launch_bounds__(256) void ntxent_main(
    const float* __restrict__ P, const int* __restrict__ ids,
    float* __restrict__ gAcc)
{
    __shared__ float aPanel[ROWS_PER_BLOCK * A_STRIDE];
    __shared__ float waveNeg[ROWS_PER_BLOCK][8];
    __shared__ float negLse[ROWS_PER_BLOCK];
    __shared__ int   rowIds[ROWS_PER_BLOCK];
    __shared__ int   hasNegS[ROWS_PER_BLOCK];
    __shared__ int   posCnt[ROWS_PER_BLOCK];
    __shared__ float lossS;

    const int t    = threadIdx.x;
    const int lane = t & 31;
    const int wave = t >> 5;
    const int r0   = blockIdx.x * ROWS_PER_BLOCK;

    // ---- stage this block's 16x256 A panel into LDS (padded stride) ----
    {
        const float4* src = (const float4*)(P + (size_t)r0 * DIM);
        #pragma unroll
        for (int q = 0; q < 4; ++q) {
            int f4  = t * 4 + q;        // 0..1023 float4s = 16 rows * 64
            int row = f4 >> 6;
            int c4  = f4 & 63;
            float4 v = src[f4];
            *(float4*)&aPanel[row * A_STRIDE + c4 * 4] = v;
        }
        if (t < ROWS_PER_BLOCK) { rowIds[t] = ids[r0 + t]; posCnt[t] = 0; }
        if (t == 0) lossS = 0.0f;
    }
    __syncthreads();

    const int l15 = lane & 15;
    const int hi  = lane >> 4;   // A/B: K offset {0,2}; C/D: row offset {0,8}

    // ids of the 8 C-matrix rows this lane's accumulator elements map to
    int myRowId[8];
    #pragma unroll
    for (int r = 0; r < 8; ++r) myRowId[r] = rowIds[r + 8 * hi];

    float negSum[8];
    #pragma unroll
    for (int r = 0; r < 8; ++r) negSum[r] = 0.0f;

    // ---- GEMM over this wave's 512-column window, fused masked exp-sum ----
    const float* aRow = aPanel + l15 * A_STRIDE + 2 * hi;   // LDS, bank-clean
    const int nBase0 = wave * 512;
    for (int tile = 0; tile < 32; ++tile) {
        const int n   = nBase0 + tile * 16 + l15;
        const int idN = ids[n];
        const float* bRow = P + (size_t)n * DIM + 2 * hi;   // B = P^T column = P row

        v8f acc = {0.f,0.f,0.f,0.f,0.f,0.f,0.f,0.f};
        #pragma unroll 16
        for (int k = 0; k < DIM; k += 4) {
            v2f a = *(const v2f*)(aRow + k);
            v2f b = *(const v2f*)(bRow + k);
            acc = __builtin_amdgcn_wmma_f32_16x16x4_f32(
                      false, a, false, b, (short)0, acc, false, false);
        }
        // epilogue: sim in [-2,2] after temperature => no max-shift needed
        #pragma unroll
        for (int r = 0; r < 8; ++r) {
            if (idN != myRowId[r]) {
                negSum[r] += __expf(acc[r] * INV_T);
            }
        }
    }

    // reduce each row's exp-sum across the 16 lanes of its half-wave (wave32)
    #pragma unroll
    for (int r = 0; r < 8; ++r) {
        float v = negSum[r];
        for (int off = 1; off < 16; off <<= 1)
            v += __shfl_xor(v, off, 32);
        negSum[r] = v;
    }
    if (l15 == 0) {
        #pragma unroll
        for (int r = 0; r < 8; ++r) waveNeg[r + 8 * hi][wave] = negSum[r];
    }
    __syncthreads();

    if (t < ROWS_PER_BLOCK) {
        float tot = 0.0f;
        #pragma unroll
        for (int w = 0; w < 8; ++w) tot += waveNeg[t][w];
        hasNegS[t] = (tot > 0.0f) ? 1 : 0;
        negLse[t]  = __logf(tot);
    }
    __syncthreads();

    // ---- phase 2: positives (rare, ~3/row) recomputed exactly ----
    float myLoss = 0.0f;
    for (int j = t; j < BATCH; j += 256) {
        const int idJ = ids[j];
        #pragma unroll 1
        for (int i = 0; i < ROWS_PER_BLOCK; ++i) {
            if (idJ == rowIds[i] && j != (r0 + i)) {
                const float* ar = aPanel + i * A_STRIDE;
                const float* pj = P + (size_t)j * DIM;
                float d = 0.0f;
                for (int k = 0; k < DIM; ++k) d = fmaf(ar[k], pj[k], d);
                const float s = d * INV_T;
                atomicAdd(&posCnt[i], 1);
                if (hasNegS[i])
                    myLoss += log1pf(__expf(negLse[i] - s));  // logaddexp(s,L)-s
            }
        }
    }
    atomicAdd(&lossS, myLoss);
    __syncthreads();

    if (t == 0) atomicAdd(&gAcc[0], lossS);
    if (t < ROWS_PER_BLOCK) {
        if (posCnt[t] > 0 && hasNegS[t])
            atomicAdd((unsigned int*)gAcc + 1, 1u);
    }
}

__global__ void ntxent_final(const float* gAcc, float* out) {
    const unsigned int cnt = ((const unsigned int*)gAcc)[1];
    const float denom = (cnt > 0u) ? (float)cnt : 1.0f;
    out[0] = (cnt > 0u) ? (gAcc[0] / denom) : 0.0f;
}

extern "C" void kernel_launch(void* const* d_in, const int* in_sizes, int n_in,
                              void* d_out, int out_size, void* d_ws, size_t ws_size,
                              hipStream_t stream) {
    const float* P   = (const float*)d_in[0];
    const int*   ids = (const int*)d_in[1];
    float* acc = (float*)d_ws;

    hipLaunchKernelGGL(ntxent_init, dim3(1), dim3(1), 0, stream, acc);
    hipLaunchKernelGGL(ntxent_main, dim3(BATCH / ROWS_PER_BLOCK), dim3(256), 0, stream,
                       P, ids, acc);
    hipLaunchKernelGGL(ntxent_final, dim3(1), dim3(1), 0, stream, acc, (float*)d_out);
}